// EulerIntegratorCell_893353198434
// MI455X (gfx1250) — compile-verified
//
#include <hip/hip_runtime.h>
#include <hip/hip_bf16.h>

typedef __attribute__((ext_vector_type(2))) float v2f;
typedef __attribute__((ext_vector_type(8))) float v8f;

namespace {
constexpr int   kB     = 16384;
constexpr int   kT     = 2048;
constexpr int   kHid   = 64;
constexpr int   kChunk = 64;           // time steps staged per LDS chunk
constexpr int   kXStr  = 65;           // padded LDS row stride (dwords) -> bank (r+s)%64
constexpr int   kWaveLds = 16 * kXStr + 32 * kXStr; // x stage + a stage (two half-wave mirrors)
constexpr float kC = 1.5e-11f;
constexpr float kM = 3.8f;
}

__device__ __forceinline__ float tanh_hw(float x) {
#if __has_builtin(__builtin_amdgcn_tanhf)
  return __builtin_amdgcn_tanhf(x);          // v_tanh_f32 (gfx1250 trans op)
#else
  float y;
  asm("v_tanh_f32 %0, %1" : "=v"(y) : "v"(x));
  return y;
#endif
}

__global__ __launch_bounds__(128) void euler_paris_wmma(
    const float* __restrict__ x, const float* __restrict__ a0,
    const float* __restrict__ W1, const float* __restrict__ b1,
    const float* __restrict__ W2, const float* __restrict__ b2,
    float* __restrict__ out) {
  __shared__ float smem[4 * kWaveLds];

  const int lane = threadIdx.x & 31;
  const int wid  = threadIdx.x >> 5;
  const int row0 = (blockIdx.x * 4 + wid) * 16;   // 16 batch rows per wave

  float* lx = smem + wid * kWaveLds;              // staged x chunk
  float* la = lx + 16 * kXStr;                    // staged a outputs

  const int  lm   = lane & 15;
  const bool lo16 = (lane < 16);
  const int  mirr = (lane >> 4) * 16 * kXStr;     // upper half writes to mirror region

  // A' tiles: W1^T with fused bias.  f32 16x4 A-layout:
  //   lanes 0-15 : VGPR0=K0, VGPR1=K1 (row n = lane)
  //   lanes 16-31: VGPR0=K2, VGPR1=K3
  v2f   atile[4];
  float w2s[4][8];
#pragma unroll
  for (int j = 0; j < 4; ++j) {
    const int n = lm + 16 * j;
    atile[j].x = lo16 ? W1[n]        : b1[n];     // K0: W1[0,n] | K2: bias (pairs with 1.0)
    atile[j].y = lo16 ? W1[kHid + n] : 0.0f;      // K1: W1[1,n] | K3: 0
#pragma unroll
    for (int v = 0; v < 8; ++v)                   // W2 slot per (tile, D-VGPR, lane-half)
      w2s[j][v] = W2[(lo16 ? v : v + 8) + 16 * j];
  }
  const float bias2 = b2[0];

  float a = a0[row0 + lm];                        // both wave halves mirror the state

  for (int c = 0; c < kT; c += kChunk) {
    // ---- stage x[row0..row0+15, c..c+63] into LDS (coalesced 128B loads) ----
#pragma unroll 4
    for (int r = 0; r < 16; ++r) {
      const float* src = x + (size_t)(row0 + r) * kT + c;
      lx[r * kXStr + lane]      = src[lane];
      lx[r * kXStr + 32 + lane] = src[32 + lane];
    }
    // ---- serial recurrence over the chunk; x_t pipelined one step ahead ----
    float xt = lx[lm * kXStr + 0];
#pragma unroll 2
    for (int s = 0; s < kChunk; ++s) {
      // issue next step's x load now; its wait lands off the critical path
      float xt_next = (s + 1 < kChunk) ? lx[lm * kXStr + (s + 1)] : 0.0f;

      v2f bmat;                                   // B' = per-row inputs {x, a, 1, 0}
      bmat.x = lo16 ? xt : 1.0f;                  // K0 | K2
      bmat.y = lo16 ? a  : 0.0f;                  // K1 | K3
      const v8f z = {0.f,0.f,0.f,0.f,0.f,0.f,0.f,0.f};
      // D'[n][m]: lane m holds h-preact of its own batch row -> no transpose needed
      v8f d0 = __builtin_amdgcn_wmma_f32_16x16x4_f32(false, atile[0], false, bmat, (short)0, z, false, false);
      v8f d1 = __builtin_amdgcn_wmma_f32_16x16x4_f32(false, atile[1], false, bmat, (short)0, z, false, false);
      v8f d2 = __builtin_amdgcn_wmma_f32_16x16x4_f32(false, atile[2], false, bmat, (short)0, z, false, false);
      v8f d3 = __builtin_amdgcn_wmma_f32_16x16x4_f32(false, atile[3], false, bmat, (short)0, z, false, false);

      // tanh + h.W2 dot: 8 independent accumulators (chain depth 4), tree combine
      float acc[8];
#pragma unroll
      for (int v = 0; v < 8; ++v)
        acc[v] = tanh_hw(d0[v]) * w2s[0][v];
#pragma unroll
      for (int v = 0; v < 8; ++v)
        acc[v] = fmaf(tanh_hw(d1[v]), w2s[1][v], acc[v]);
#pragma unroll
      for (int v = 0; v < 8; ++v)
        acc[v] = fmaf(tanh_hw(d2[v]), w2s[2][v], acc[v]);
#pragma unroll
      for (int v = 0; v < 8; ++v)
        acc[v] = fmaf(tanh_hw(d3[v]), w2s[3][v], acc[v]);
      const float p = ((acc[0] + acc[1]) + (acc[2] + acc[3])) +
                      ((acc[4] + acc[5]) + (acc[6] + acc[7]));

      // single cross-half combine: lane m + lane m+16 hold the two 32-slot halves
      const float dk = (p + __shfl_xor(p, 16, 32)) + bias2;
      // a += C * dk^3.8  via v_log_f32 / v_exp_f32 (dk > 0 by construction)
      const float l2 = __builtin_amdgcn_logf(dk);
      a = fmaf(kC, __builtin_amdgcn_exp2f(kM * l2), a);
      la[mirr + lm * kXStr + s] = a;              // halves write disjoint banks

      xt = xt_next;
    }
    // ---- coalesced writeback of the a_seq chunk ----
#pragma unroll 4
    for (int r = 0; r < 16; ++r) {
      float* dst = out + (size_t)(row0 + r) * kT + c;
      dst[lane]      = la[r * kXStr + lane];
      dst[32 + lane] = la[r * kXStr + 32 + lane];
    }
  }
}

extern "C" void kernel_launch(void* const* d_in, const int* in_sizes, int n_in,
                              void* d_out, int out_size, void* d_ws, size_t ws_size,
                              hipStream_t stream) {
  (void)in_sizes; (void)n_in; (void)out_size; (void)d_ws; (void)ws_size;
  const float* x  = (const float*)d_in[0];
  const float* a0 = (const float*)d_in[1];
  const float* W1 = (const float*)d_in[2];
  const float* b1 = (const float*)d_in[3];
  const float* W2 = (const float*)d_in[4];
  const float* b2 = (const float*)d_in[5];
  float* out = (float*)d_out;
  dim3 grid(kB / 16 / 4);   // 256 blocks x 4 waves = 1024 wave-tiles of 16 rows
  dim3 block(128);
  euler_paris_wmma<<<grid, block, 0, stream>>>(x, a0, W1, b1, W2, b2, out);
}